// CustomBCELoss_58737972740371
// MI455X (gfx1250) — compile-verified
//
#include <hip/hip_runtime.h>

typedef float f32x4 __attribute__((ext_vector_type(4)));
typedef float v2f   __attribute__((ext_vector_type(2)));
typedef float v8f   __attribute__((ext_vector_type(8)));

#define SPATIAL   (128*128*128)   // 2,097,152 elements per (n,c) chunk
#define NCHUNK    12              // N(4) * C(3) contiguous chunks
#define NCH       3
#define BPC       128             // blocks per chunk
#define TPB       256
#define NBLK      (BPC*NCHUNK)    // 1536 partial slots per quantity
#define LOG_CLAMP (-100.0f)
#define EMPTY_W   (1000.0f)

// Pass 1: streaming BCE partial sums. Channel is uniform per chunk, so each
// block's partials map to a known channel via chunk = blockIdx.y.
__global__ __launch_bounds__(TPB) void bce_pass1(const float* __restrict__ p,
                                                 const float* __restrict__ t,
                                                 float* __restrict__ partials) {
    const int chunk = blockIdx.y;
    const f32x4* p4 = (const f32x4*)(p + (size_t)chunk * SPATIAL);
    const f32x4* t4 = (const f32x4*)(t + (size_t)chunk * SPATIAL);
    const int nvec = SPATIAL / 4;   // 524288 float4 per chunk

    float accB = 0.0f;  // sum of t*log(p)+(1-t)*log(1-p), clamped
    float accO = 0.0f;  // count of t != 0

    for (int i = blockIdx.x * TPB + threadIdx.x; i < nvec; i += gridDim.x * TPB) {
        f32x4 pv = __builtin_nontemporal_load(&p4[i]);   // global_load_b128, TH=NT
        f32x4 tv = __builtin_nontemporal_load(&t4[i]);
        #pragma unroll
        for (int j = 0; j < 4; ++j) {
            float pe = pv[j];
            float te = tv[j];
            float lp = fmaxf(__logf(pe), LOG_CLAMP);          // v_log_f32 path
            float l1 = fmaxf(__logf(1.0f - pe), LOG_CLAMP);
            accB = fmaf(te, lp, accB);
            accB = fmaf(1.0f - te, l1, accB);
            accO += (te != 0.0f) ? 1.0f : 0.0f;
        }
    }

    __shared__ float sB[TPB];
    __shared__ float sO[TPB];
    sB[threadIdx.x] = accB;
    sO[threadIdx.x] = accO;
    __syncthreads();
    #pragma unroll
    for (int s = TPB / 2; s > 0; s >>= 1) {
        if (threadIdx.x < s) {
            sB[threadIdx.x] += sB[threadIdx.x + s];
            sO[threadIdx.x] += sO[threadIdx.x + s];
        }
        __syncthreads();
    }
    if (threadIdx.x == 0) {
        int b = chunk * gridDim.x + blockIdx.x;
        partials[b]        = sB[0];
        partials[NBLK + b] = sO[0];
    }
}

// Pass 2: deterministic fixed-order reduction of 1536 partials per quantity,
// per-channel weight/bce, then the final weighted mean as a K=3 dot product
// via V_WMMA_F32_16X16X4_F32 (A row0 = w/3, B col0 = bce; D[0][0] = answer).
__global__ __launch_bounds__(TPB) void bce_pass2(const float* __restrict__ partials,
                                                 float* __restrict__ out) {
    float cb[NCH] = {0.f, 0.f, 0.f};
    float co[NCH] = {0.f, 0.f, 0.f};
    for (int i = threadIdx.x; i < NBLK; i += TPB) {
        int c = (i / BPC) % NCH;   // chunk % 3
        cb[c] += partials[i];
        co[c] += partials[NBLK + i];
    }

    __shared__ float sb[NCH][TPB];
    __shared__ float so[NCH][TPB];
    #pragma unroll
    for (int c = 0; c < NCH; ++c) {
        sb[c][threadIdx.x] = cb[c];
        so[c][threadIdx.x] = co[c];
    }
    __syncthreads();
    for (int s = TPB / 2; s > 0; s >>= 1) {
        if (threadIdx.x < s) {
            #pragma unroll
            for (int c = 0; c < NCH; ++c) {
                sb[c][threadIdx.x] += sb[c][threadIdx.x + s];
                so[c][threadIdx.x] += so[c][threadIdx.x + s];
            }
        }
        __syncthreads();
    }

    const float M = (float)(4 * SPATIAL);  // 8,388,608 elements per channel row
    float w[NCH], b[NCH];
    #pragma unroll
    for (int c = 0; c < NCH; ++c) {
        float ones = so[c][0];
        w[c] = (ones > 0.0f) ? (M / fmaxf(ones, 1.0f)) : EMPTY_W;
        b[c] = -sb[c][0] / M;
    }

    // A (16x4 f32): lanes 0-15 hold rows; VGPR0 = {K0 | K2}, VGPR1 = {K1 | K3}.
    // Row 0 = [w0/3, w1/3, w2/3, 0] -> lane0:{A.x=w0/3, A.y=w1/3}, lane16:{A.x=w2/3}.
    // B (4x16 f32) mirrored: col 0 = [b0, b1, b2, 0].
    const int lane = threadIdx.x & 31;   // wave32
    const float third = 1.0f / 3.0f;
    v2f A = {0.0f, 0.0f};
    v2f B = {0.0f, 0.0f};
    if (lane == 0)  { A[0] = w[0] * third; A[1] = w[1] * third; B[0] = b[0]; B[1] = b[1]; }
    if (lane == 16) { A[0] = w[2] * third; B[0] = b[2]; }
    v8f Cm = {};
    v8f D = __builtin_amdgcn_wmma_f32_16x16x4_f32(
        /*neg_a=*/false, A, /*neg_b=*/false, B,
        /*c_mod=*/(short)0, Cm, /*reuse_a=*/false, /*reuse_b=*/false);

    if (threadIdx.x == 0) out[0] = D[0];   // D[M=0][N=0] lives in VGPR0, lane 0
}

extern "C" void kernel_launch(void* const* d_in, const int* in_sizes, int n_in,
                              void* d_out, int out_size, void* d_ws, size_t ws_size,
                              hipStream_t stream) {
    const float* input  = (const float*)d_in[0];
    const float* target = (const float*)d_in[1];
    float* partials = (float*)d_ws;       // 2 * 1536 floats = 12 KB scratch
    float* out = (float*)d_out;

    dim3 grid1(BPC, NCHUNK);              // 1536 blocks x 256 thr = 12288 waves
    bce_pass1<<<grid1, TPB, 0, stream>>>(input, target, partials);
    bce_pass2<<<1, TPB, 0, stream>>>(partials, out);
}